// ResonantLayer_75892072121068
// MI455X (gfx1250) — compile-verified
//
#include <hip/hip_runtime.h>
#include <hip/hip_bf16.h>
#include <math.h>

// Problem sizes (fixed by the reference)
#define B_SZ    2048
#define D_MODEL 512
#define N_NEUR  128
#define LUT_SZ  4096
#define LUT_MASK 4095
// LUT_SIZE / (2*pi)
#define LUT_SCALE 651.8986469044033f

typedef __attribute__((ext_vector_type(2))) float v2f;
typedef __attribute__((ext_vector_type(8))) float v8f;
typedef __attribute__((ext_vector_type(4))) int   v4i;

#define AS1 __attribute__((address_space(1)))
#define AS3 __attribute__((address_space(3)))

// CDNA5 async global->LDS copy (ASYNCcnt-tracked), with sync fallback.
#if defined(__has_builtin)
# if __has_builtin(__builtin_amdgcn_global_load_async_to_lds_b128)
#  define HAVE_ASYNC_LDS 1
# endif
#endif
#ifndef HAVE_ASYNC_LDS
# define HAVE_ASYNC_LDS 0
#endif

__device__ __forceinline__ void wait_asynccnt_zero() {
#if defined(__has_builtin) && __has_builtin(__builtin_amdgcn_s_wait_asynccnt)
    __builtin_amdgcn_s_wait_asynccnt(0);
#else
    asm volatile("s_wait_asynccnt 0x0" ::: "memory");
#endif
}

// ---------------------------------------------------------------------------
// Kernel 0a: fold the LUT scale into per-(n,d) constants, interleaved so the
// hot loop needs one vector load per element pair:
//   wp[n,d] = { LUT_SCALE / (1 + |W[n,d]|),  LUT_SCALE * Bp[n,d] }
// ---------------------------------------------------------------------------
__global__ void prep_kernel(const float* __restrict__ W,
                            const float* __restrict__ Bp,
                            float2* __restrict__ wp) {
    int i = blockIdx.x * blockDim.x + threadIdx.x;
    if (i < N_NEUR * D_MODEL) {
        wp[i] = make_float2(LUT_SCALE / (1.0f + fabsf(W[i])),
                            LUT_SCALE * Bp[i]);
    }
}

// ---------------------------------------------------------------------------
// Kernel 0b: build the interleaved {cos,sin} table once in global memory
// (32 KB -> lives in L2 for all 2048 consumer blocks) so each block can
// async-DMA it straight into LDS in b128 chunks.
// ---------------------------------------------------------------------------
__global__ void tab_prep_kernel(const float* __restrict__ sin_t,
                                const float* __restrict__ cos_t,
                                float2* __restrict__ ctab) {
    int i = blockIdx.x * blockDim.x + threadIdx.x;
    if (i < LUT_SZ) ctab[i] = make_float2(cos_t[i], sin_t[i]);
}

// ---------------------------------------------------------------------------
// Kernel 1: x_collapsed = [x_real | x_imag] (2048x1024) @ Wc^T (1024x512) + bc
// One 16x16 f32 tile per wave via V_WMMA_F32_16X16X4_F32; K-loop split into
// the x_real half and x_imag half; A/B fragments are single b64 loads.
// Fragment layouts per CDNA5 ISA 7.12.2.
// ---------------------------------------------------------------------------
__global__ void collapse_gemm_kernel(const float* __restrict__ xr,
                                     const float* __restrict__ xi,
                                     const float* __restrict__ Wc,
                                     const float* __restrict__ bc,
                                     float* __restrict__ xc) {
    const int wave = threadIdx.x >> 5;
    const int lane = threadIdx.x & 31;
    const int half = lane >> 4;        // 0: K+0..1, 1: K+2..3
    const int l    = lane & 15;

    const int tile = blockIdx.x * 8 + wave;   // 128 M-tiles x 32 N-tiles
    const int tm = tile >> 5;
    const int tn = tile & 31;
    const int m0 = tm * 16;
    const int n0 = tn * 16;

    const float* __restrict__ ar = xr + (m0 + l) * D_MODEL + half * 2;
    const float* __restrict__ ai = xi + (m0 + l) * D_MODEL + half * 2;
    const float* __restrict__ bp = Wc + (n0 + l) * (2 * D_MODEL) + half * 2;

    v8f acc = {};
    #pragma unroll 8
    for (int k = 0; k < D_MODEL; k += 4) {          // K half 1: x_real
        v2f a = *(const v2f*)(ar + k);
        v2f b = *(const v2f*)(bp + k);
        acc = __builtin_amdgcn_wmma_f32_16x16x4_f32(
                  false, a, false, b, (short)0, acc, false, false);
    }
    #pragma unroll 8
    for (int k = 0; k < D_MODEL; k += 4) {          // K half 2: x_imag
        v2f a = *(const v2f*)(ai + k);
        v2f b = *(const v2f*)(bp + D_MODEL + k);
        acc = __builtin_amdgcn_wmma_f32_16x16x4_f32(
                  false, a, false, b, (short)0, acc, false, false);
    }

    const float bias = bc[n0 + l];
    #pragma unroll
    for (int r = 0; r < 8; ++r) {
        const int row = m0 + r + half * 8;
        xc[row * D_MODEL + (n0 + l)] = acc[r] + bias;
    }
}

// ---------------------------------------------------------------------------
// Kernel 2 (dominant, 134M elements): for each (b,n):
//   pos  = xc[b,d]*wp.x + wp.y + t[b]*SCALE            (LUT coordinates)
//   i0   = (int)floor(pos) & 4095 ; frac = pos - floor(pos)
//   lerp cos/sin from interleaved float2 LDS table (1 ds_load_b64 per index)
//   reduce over d (512) -> cosS[b,n], sinS[b,n]
// One b per block (2048 blocks), 8 waves; wave w owns n = w + 8*j.
// Each lane handles TWO consecutive d per iteration: one global_load_b128
// (wp pair) + one ds b64 (x pair) + two table lerps.
// LDS staging (32KB table + 2KB x row) via GLOBAL_LOAD_ASYNC_TO_LDS_B128.
// ---------------------------------------------------------------------------
__global__ void lut_reduce_kernel(const float* __restrict__ xc,
                                  const float2* __restrict__ wp,
                                  const float* __restrict__ t,
                                  const float2* __restrict__ ctab,
                                  float* __restrict__ cosS,
                                  float* __restrict__ sinS) {
    __shared__ __align__(16) float2 tab[LUT_SZ];     // .x=cos .y=sin (32 KB)
    __shared__ __align__(16) float  xrow[D_MODEL];   // 2 KB

    const int b = blockIdx.x;

#if HAVE_ASYNC_LDS
    {
        // 32 KB table: 2048 x 16B async transfers
        AS1 v4i* gt = (AS1 v4i*)const_cast<float2*>(ctab);
        AS3 v4i* lt = (AS3 v4i*)tab;
        for (int i = threadIdx.x; i < (LUT_SZ * 8) / 16; i += blockDim.x)
            __builtin_amdgcn_global_load_async_to_lds_b128(gt + i, lt + i, 0, 0);
        // 2 KB x row: 128 x 16B async transfers
        AS1 v4i* gx = (AS1 v4i*)const_cast<float*>(xc + (size_t)b * D_MODEL);
        AS3 v4i* lx = (AS3 v4i*)xrow;
        for (int i = threadIdx.x; i < D_MODEL / 4; i += blockDim.x)
            __builtin_amdgcn_global_load_async_to_lds_b128(gx + i, lx + i, 0, 0);
    }
    wait_asynccnt_zero();
#else
    for (int i = threadIdx.x; i < LUT_SZ; i += blockDim.x)
        tab[i] = ctab[i];
    for (int i = threadIdx.x; i < D_MODEL; i += blockDim.x)
        xrow[i] = xc[(size_t)b * D_MODEL + i];
#endif
    __syncthreads();

    const float tb = t[b] * LUT_SCALE;
    const int wave = threadIdx.x >> 5;
    const int lane = threadIdx.x & 31;

    for (int n = wave; n < N_NEUR; n += 8) {
        const float2* __restrict__ w = wp + n * D_MODEL;
        float cs = 0.0f, ss = 0.0f;
        #pragma unroll 4
        for (int d = lane * 2; d < D_MODEL; d += 64) {
            const float4 c  = *(const float4*)(w + d);     // {rwl0,bp0,rwl1,bp1}
            const float2 xv = *(const float2*)(xrow + d);  // ds_load_b64

            // element 0
            float pos  = fmaf(xv.x, c.x, c.y + tb);
            float base = floorf(pos);
            float frac = pos - base;
            int  i0 = ((int)base) & LUT_MASK;              // == jnp.mod(.,4096)
            int  i1 = (i0 + 1) & LUT_MASK;
            float2 v0 = tab[i0];
            float2 v1 = tab[i1];
            cs += fmaf(frac, v1.x - v0.x, v0.x);
            ss += fmaf(frac, v1.y - v0.y, v0.y);

            // element 1
            pos  = fmaf(xv.y, c.z, c.w + tb);
            base = floorf(pos);
            frac = pos - base;
            i0 = ((int)base) & LUT_MASK;
            i1 = (i0 + 1) & LUT_MASK;
            v0 = tab[i0];
            v1 = tab[i1];
            cs += fmaf(frac, v1.x - v0.x, v0.x);
            ss += fmaf(frac, v1.y - v0.y, v0.y);
        }
        // wave32 butterfly reduction
        #pragma unroll
        for (int off = 16; off > 0; off >>= 1) {
            cs += __shfl_xor(cs, off, 32);
            ss += __shfl_xor(ss, off, 32);
        }
        if (lane == 0) {
            cosS[b * N_NEUR + n] = cs;
            sinS[b * N_NEUR + n] = ss;
        }
    }
}

// ---------------------------------------------------------------------------
// Kernel 3: out (2048x512) = S (2048x128) @ Wm^T, Wm (512x128) row-major.
// Same WMMA tiling, K = 128 (32 wmma steps), b64 fragment loads.
// ---------------------------------------------------------------------------
__global__ void out_gemm_kernel(const float* __restrict__ S,
                                const float* __restrict__ Wm,
                                float* __restrict__ out) {
    const int wave = threadIdx.x >> 5;
    const int lane = threadIdx.x & 31;
    const int half = lane >> 4;
    const int l    = lane & 15;

    const int tile = blockIdx.x * 8 + wave;   // 128 x 32 tiles
    const int tm = tile >> 5;
    const int tn = tile & 31;
    const int m0 = tm * 16;
    const int n0 = tn * 16;

    const float* __restrict__ ap = S  + (m0 + l) * N_NEUR + half * 2;
    const float* __restrict__ bp = Wm + (n0 + l) * N_NEUR + half * 2;

    v8f acc = {};
    #pragma unroll 8
    for (int k = 0; k < N_NEUR; k += 4) {
        v2f a = *(const v2f*)(ap + k);
        v2f b = *(const v2f*)(bp + k);
        acc = __builtin_amdgcn_wmma_f32_16x16x4_f32(
                  false, a, false, b, (short)0, acc, false, false);
    }

    #pragma unroll
    for (int r = 0; r < 8; ++r) {
        const int row = m0 + r + half * 8;
        out[row * D_MODEL + (n0 + l)] = acc[r];
    }
}

// ---------------------------------------------------------------------------
// Launch. Inputs (setup_inputs order):
//  0 x_real (B,D) 1 x_imag (B,D) 2 t (B) 3 Wc (D,2D) 4 bc (D)
//  5 W (N,D) 6 Bp (N,D) 7 Wr (D,N) 8 Wi (D,N) 9 sin_table 10 cos_table
// Output: out_real (B,D) then out_imag (B,D), fp32, concatenated.
// ---------------------------------------------------------------------------
extern "C" void kernel_launch(void* const* d_in, const int* in_sizes, int n_in,
                              void* d_out, int out_size, void* d_ws, size_t ws_size,
                              hipStream_t stream) {
    const float* xr    = (const float*)d_in[0];
    const float* xi    = (const float*)d_in[1];
    const float* t     = (const float*)d_in[2];
    const float* Wc    = (const float*)d_in[3];
    const float* bc    = (const float*)d_in[4];
    const float* W     = (const float*)d_in[5];
    const float* Bp    = (const float*)d_in[6];
    const float* Wr    = (const float*)d_in[7];
    const float* Wi    = (const float*)d_in[8];
    const float* sin_t = (const float*)d_in[9];
    const float* cos_t = (const float*)d_in[10];

    float* out_real = (float*)d_out;
    float* out_imag = out_real + (size_t)B_SZ * D_MODEL;

    // Workspace layout
    float*  xc   = (float*)d_ws;                                // B*D floats
    float2* wp   = (float2*)(xc + (size_t)B_SZ * D_MODEL);      // N*D float2
    float*  cosS = (float*)(wp + (size_t)N_NEUR * D_MODEL);     // B*N floats
    float*  sinS = cosS + (size_t)B_SZ * N_NEUR;                // B*N floats
    float2* ctab = (float2*)(sinS + (size_t)B_SZ * N_NEUR);     // LUT_SZ float2

    // 0) fold LUT scale into interleaved constants + build combined table
    prep_kernel<<<(N_NEUR * D_MODEL + 255) / 256, 256, 0, stream>>>(W, Bp, wp);
    tab_prep_kernel<<<(LUT_SZ + 255) / 256, 256, 0, stream>>>(sin_t, cos_t, ctab);

    // 1) input collapse GEMM (WMMA f32): 4096 tiles / 8 waves = 512 blocks
    collapse_gemm_kernel<<<512, 256, 0, stream>>>(xr, xi, Wc, bc, xc);

    // 2) LUT sin/cos + D-reduction: one b per block
    lut_reduce_kernel<<<B_SZ, 256, 0, stream>>>(xc, wp, t, ctab, cosS, sinS);

    // 3) output GEMMs (WMMA f32)
    out_gemm_kernel<<<512, 256, 0, stream>>>(cosS, Wr, out_real);
    out_gemm_kernel<<<512, 256, 0, stream>>>(sinS, Wi, out_imag);
}